// RNNLayer_48258252538205
// MI455X (gfx1250) — compile-verified
//
#include <hip/hip_runtime.h>
#include <hip/hip_bf16.h>

// Reference:  r = tanh(y);  dy = (-y + W@r + x)/TAU;  out = y + DT*dy
// N = 8192, W is fp32 NxN (256 MB) -> pure HBM-bandwidth GEMV.
// inputs: d_in[0]=x (N f32), d_in[1]=y (N f32), d_in[2]=W (N*N f32)

#define NN 8192
#define DT 1e-3f
#define TAU 1e-2f
#define KSPLIT 16              // K-dimension split for occupancy
#define KCHUNK (NN / KSPLIT)   // 512 columns per wave
#define ROWTILES (NN / 16)     // 512 row tiles
#define WAVES_PER_BLOCK 8      // 256 threads = 8 wave32

typedef __attribute__((ext_vector_type(2))) float v2f;
typedef __attribute__((ext_vector_type(8))) float v8f;

// ---------------------------------------------------------------------------
// Kernel 1: r = tanh(y), acc = 0  (workspace is poisoned each timing run)
// ---------------------------------------------------------------------------
__global__ void rnn_prep_kernel(const float* __restrict__ y,
                                float* __restrict__ r,
                                float* __restrict__ acc, int n) {
  int i = blockIdx.x * blockDim.x + threadIdx.x;
  if (i < n) {
    r[i]   = tanhf(y[i]);
    acc[i] = 0.0f;
  }
}

// ---------------------------------------------------------------------------
// Kernel 2: acc += W[tile, kchunk] @ r[kchunk]  via V_WMMA_F32_16X16X4_F32
//
// A-operand ISA layout (16x4 f32): lane l (0..15) holds row M=l, K=0,1 in its
// two VGPRs; lanes 16..31 hold row M=l-16, K=2,3.  -> one global_load_b64 of
// W[row0 + (l&15)][k + 2*(l>>4)] feeds a whole WMMA.
// B-operand (4x16 f32): row K striped across lanes within a VGPR; we broadcast
// r[k+kk] into every column, so D[m, n] == dot(W_row_m, r_chunk) for all n.
// D layout: VGPR v, lanes 0-15: M=v; lanes 16-31: M=v+8 -> lane 0 / lane 16
// hold the 16 row results and flush them with float atomics.
// ---------------------------------------------------------------------------
__global__ __launch_bounds__(256) void
rnn_gemv_wmma_kernel(const float* __restrict__ W,
                     const float* __restrict__ r,
                     float* __restrict__ acc) {
  const int lane  = threadIdx.x & 31;
  const int wave  = threadIdx.x >> 5;
  const int gwave = blockIdx.x * WAVES_PER_BLOCK + wave;

  const int rowTile = gwave % ROWTILES;   // which 16 output rows
  const int kChunk  = gwave / ROWTILES;   // which 512-column slice of K
  const int row0    = rowTile * 16;
  const int k0      = kChunk * KCHUNK;

  const int halfSel = lane >> 4;          // 0: K=0,1   1: K=2,3
  const int m       = lane & 15;          // row within tile (A layout)

  const float* wp = W + (size_t)(row0 + m) * NN + (k0 + 2 * halfSel);
  const float* rp = r + (k0 + 2 * halfSel);

  v8f c = {};
#pragma unroll 4
  for (int kk = 0; kk < KCHUNK; kk += 4) {
    v2f a = *(const v2f*)wp;              // W[row][k+2h], W[row][k+2h+1]
    v2f b = *(const v2f*)rp;              // r[k+2h], r[k+2h+1] (bcast per half)
    __builtin_prefetch(wp + 512, 0, 1);   // stream W ~2KB ahead per lane
    // (neg_a, A, neg_b, B, c_mod, C, reuse_a, reuse_b)
    c = __builtin_amdgcn_wmma_f32_16x16x4_f32(
        false, a, false, b, (short)0, c, false, false);
    wp += 4;
    rp += 4;
  }

  // Every D column is identical; lane 0 holds rows 0..7, lane 16 rows 8..15.
  if ((lane & 15) == 0) {
    const int base = row0 + (halfSel ? 8 : 0);
#pragma unroll
    for (int i = 0; i < 8; ++i) {
      unsafeAtomicAdd(&acc[base + i], c[i]);   // global_atomic_add_f32
    }
  }
}

// ---------------------------------------------------------------------------
// Kernel 3: out = y + DT * ((-y + acc + x) / TAU)
// ---------------------------------------------------------------------------
__global__ void rnn_finish_kernel(const float* __restrict__ x,
                                  const float* __restrict__ y,
                                  const float* __restrict__ acc,
                                  float* __restrict__ out, int n) {
  int i = blockIdx.x * blockDim.x + threadIdx.x;
  if (i < n) {
    float dy = (-y[i] + acc[i] + x[i]) * (1.0f / TAU);
    out[i] = y[i] + DT * dy;
  }
}

extern "C" void kernel_launch(void* const* d_in, const int* in_sizes, int n_in,
                              void* d_out, int out_size, void* d_ws, size_t ws_size,
                              hipStream_t stream) {
  const float* x = (const float*)d_in[0];
  const float* y = (const float*)d_in[1];
  const float* W = (const float*)d_in[2];
  float* out = (float*)d_out;

  float* acc = (float*)d_ws;        // N floats of partial sums
  float* r   = acc + NN;            // N floats of tanh(y)

  rnn_prep_kernel<<<NN / 256, 256, 0, stream>>>(y, r, acc, NN);

  const int totalWaves = ROWTILES * KSPLIT;            // 8192 waves
  const int blocks     = totalWaves / WAVES_PER_BLOCK; // 1024 blocks
  rnn_gemv_wmma_kernel<<<blocks, 256, 0, stream>>>(W, r, acc);

  rnn_finish_kernel<<<NN / 256, 256, 0, stream>>>(x, y, acc, out, NN);
}